// GetDensity_19301583028807
// MI455X (gfx1250) — compile-verified
//
#include <hip/hip_runtime.h>
#include <hip/hip_bf16.h>

typedef float v8f __attribute__((ext_vector_type(8)));
typedef float v2f __attribute__((ext_vector_type(2)));

static __device__ __constant__ float PI_OVER_CUT = 0.628318530717958647692f; // pi / 5.0

#define TSTR 68   // activation tile row stride (floats), padded: bank- and b64-friendly
#define WSTR 66   // transposed weight row stride (floats), even => 8B-aligned pairs

__device__ __forceinline__ int imin(int a, int b) { return a < b ? a : b; }

// ---------------------------------------------------------------------------
// Parameter flat layouts (pytree leaf order, row-major):
// emb MLP (din=1, nl=[1,8], dout=D):
//   w0@0(1) b0@1 g0@2 be0@3 | w1@4(8) b1@12 g1@20 be1@28 | w_out@36(8*D) b_out@36+8*D
// big MLP (din=64, nl=[64,64], dout=D):
//   w0@0(4096) b0@4096 g0@4160 be0@4224 | w1@4288(4096) b1@8384 g1@8448 be1@8512
//   w_out@8576(64*D) b_out@8576+64*D    (total per oc MLP = 9096)
// ---------------------------------------------------------------------------

// Edge-embedding (24) and center-coeff (64) constant vectors.
// (width-1 first hidden layer => layernorm output == be exactly, so these are
//  input-independent constants.)
__global__ void nn_consts_kernel(const float* __restrict__ embN,
                                 const float* __restrict__ embC,
                                 float* __restrict__ out) {
  if (threadIdx.x != 0 || blockIdx.x != 0) return;
  {
    float be0 = embN[3];
    float h0 = be0 / (1.f + expf(-be0));           // silu(be0)
    float z[8]; float mu = 0.f;
    for (int j = 0; j < 8; ++j) { z[j] = h0 * embN[4 + j] + embN[12 + j]; mu += z[j]; }
    mu *= 0.125f;
    float vr = 0.f;
    for (int j = 0; j < 8; ++j) { float d = z[j] - mu; vr += d * d; }
    vr *= 0.125f;
    float rstd = rsqrtf(vr + 1e-5f);
    float hv[8];
    for (int j = 0; j < 8; ++j) {
      float x = (z[j] - mu) * rstd * embN[20 + j] + embN[28 + j];
      hv[j] = x / (1.f + expf(-x));
    }
    for (int n = 0; n < 24; ++n) {
      float acc = embN[228 + n];
      for (int k = 0; k < 8; ++k) acc += hv[k] * embN[36 + k * 24 + n];
      out[n] = acc;
    }
  }
  {
    float be0 = embC[3];
    float h0 = be0 / (1.f + expf(-be0));
    float z[8]; float mu = 0.f;
    for (int j = 0; j < 8; ++j) { z[j] = h0 * embC[4 + j] + embC[12 + j]; mu += z[j]; }
    mu *= 0.125f;
    float vr = 0.f;
    for (int j = 0; j < 8; ++j) { float d = z[j] - mu; vr += d * d; }
    vr *= 0.125f;
    float rstd = rsqrtf(vr + 1e-5f);
    float hv[8];
    for (int j = 0; j < 8; ++j) {
      float x = (z[j] - mu) * rstd * embC[20 + j] + embC[28 + j];
      hv[j] = x / (1.f + expf(-x));
    }
    for (int m = 0; m < 64; ++m) {
      float acc = embC[548 + m];
      for (int k = 0; k < 8; ++k) acc += hv[k] * embC[36 + k * 64 + m];
      out[32 + m] = acc;
    }
  }
}

// Edge setup: dist_vec (-> d_out), cut/radial, weighted orbital scatter-add
// into center_orbital[n_atom][4][8] (6.4 MB accumulator -> lives in L2).
__global__ __launch_bounds__(256)
void edge_setup_kernel(const float* __restrict__ cart, const int* __restrict__ nl,
                       const float* __restrict__ shifts, const float* __restrict__ consts,
                       float* __restrict__ dvec_out, float* __restrict__ center_orb,
                       int n_edge) {
  int e = blockIdx.x * 256 + threadIdx.x;
  if (e >= n_edge) return;
  int i0 = nl[e];
  int j1 = nl[n_edge + e];
  float dx = cart[i0 * 3 + 0] - cart[j1 * 3 + 0] - shifts[e * 3 + 0];
  float dy = cart[i0 * 3 + 1] - cart[j1 * 3 + 1] - shifts[e * 3 + 1];
  float dz = cart[i0 * 3 + 2] - cart[j1 * 3 + 2] - shifts[e * 3 + 2];
  dvec_out[e * 3 + 0] = dx;
  dvec_out[e * 3 + 1] = dy;
  dvec_out[e * 3 + 2] = dz;
  float dist = sqrtf(dx * dx + dy * dy + dz * dz);
  float cw = 0.5f * cosf(dist * PI_OVER_CUT) + 0.5f;
  float cut = cw * cw;
  float ang1 = cut * dx, ang2 = cut * dy, ang3 = cut * dz;
  float* co = center_orb + (size_t)i0 * 32;
#pragma unroll
  for (int k = 0; k < 8; ++k) {
    float t = consts[8 + k] * (dist - consts[16 + k]);   // alpha*(d - rs)
    float wr = consts[k] * expf(-t * t);                 // w8[k] * radial[k]
    atomicAdd(&co[0 * 8 + k], cut * wr);
    atomicAdd(&co[1 * 8 + k], ang1 * wr);
    atomicAdd(&co[2 * 8 + k], ang2 * wr);
    atomicAdd(&co[3 * 8 + k], ang3 * wr);
  }
}

// OC message pass: w = iter_coeff[j1]*orbital + center_old[j1]*cut, scattered
// to center_new[i0]. orbital recomputed from dist_vec (already in d_out) --
// trades ~160 B/edge of HBM traffic per loop for a few transcendentals.
__global__ __launch_bounds__(256)
void oc_edge_kernel(const float* __restrict__ dvec, const int* __restrict__ nl,
                    const float* __restrict__ consts, const float* __restrict__ iter_coeff,
                    const float* __restrict__ co_old, float* __restrict__ co_new,
                    int n_edge) {
  int e = blockIdx.x * 256 + threadIdx.x;
  if (e >= n_edge) return;
  int i0 = nl[e];
  int j1 = nl[n_edge + e];
  float dx = dvec[e * 3 + 0], dy = dvec[e * 3 + 1], dz = dvec[e * 3 + 2];
  float dist = sqrtf(dx * dx + dy * dy + dz * dz);
  float cw = 0.5f * cosf(dist * PI_OVER_CUT) + 0.5f;
  float cut = cw * cw;
  float ang[4] = {cut, cut * dx, cut * dy, cut * dz};
  float rad[8];
#pragma unroll
  for (int k = 0; k < 8; ++k) {
    float t = consts[8 + k] * (dist - consts[16 + k]);
    rad[k] = expf(-t * t);
  }
  const float* icp = iter_coeff + (size_t)j1 * 8;
  const float* old = co_old + (size_t)j1 * 32;
  float* dst = co_new + (size_t)i0 * 32;
#pragma unroll
  for (int j = 0; j < 4; ++j) {
#pragma unroll
    for (int k = 0; k < 8; ++k) {
      float w = icp[k] * ang[j] * rad[k] + old[j * 8 + k] * cut;
      atomicAdd(&dst[j * 8 + k], w);
    }
  }
}

// density[a][m] = cc64[m] * sum_j ( sum_k CO[a][j][k] * cc[ipsin(j)][k][m] )^2
__global__ __launch_bounds__(256)
void density_kernel(const float* __restrict__ center_orb, const float* __restrict__ ccoef,
                    const float* __restrict__ cc64, float* __restrict__ density, int n_atom) {
  int idx = blockIdx.x * 256 + threadIdx.x;
  int a = idx >> 6;
  int m = idx & 63;
  if (a >= n_atom) return;
  const float* co = center_orb + (size_t)a * 32;
  float dens = 0.f;
#pragma unroll
  for (int j = 0; j < 4; ++j) {
    const float* c = ccoef + (j == 0 ? 0 : 512) + m;
    float s = 0.f;
#pragma unroll
    for (int k = 0; k < 8; ++k) s += co[j * 8 + k] * c[k * 64];
    dens += s * s;
  }
  density[(size_t)a * 64 + m] = dens * cc64[m];
}

// ------------------------- WMMA f32 MLP (64->64->64->dout) ------------------
// One wave owns a 16-row tile held in a wave-private LDS buffer (in-place
// across layers: per-wave DS ops are in-order, and every A-fragment load
// precedes every D store in program order). Weights are staged once per
// block into LDS, transposed so B fragments are single ds_load_b64 reads.

__device__ __forceinline__ v8f wmma_f32(v2f a, v2f b, v8f c) {
  return __builtin_amdgcn_wmma_f32_16x16x4_f32(false, a, false, b, (short)0, c, false, false);
}

// D[16][TSTR](lds) = A[16][TSTR](lds) @ Wt^T + bias   (A and D may alias)
__device__ __forceinline__ void wave_gemm_lds(const float* A, const float* Wt,
                                              const float* bias, float* D, unsigned lane) {
  const int mr = lane & 15;
  const int hi = lane >> 4;
  v8f acc[4];
#pragma unroll
  for (int t = 0; t < 4; ++t) acc[t] = (v8f){};
#pragma unroll
  for (int k0 = 0; k0 < 64; k0 += 4) {
    const int kb = k0 + 2 * hi;
    v2f a = *(const v2f*)(A + mr * TSTR + kb);
#pragma unroll
    for (int t = 0; t < 4; ++t) {
      v2f b = *(const v2f*)(Wt + (t * 16 + mr) * WSTR + kb);
      acc[t] = wmma_f32(a, b, acc[t]);
    }
  }
  __builtin_amdgcn_wave_barrier();
#pragma unroll
  for (int t = 0; t < 4; ++t)
#pragma unroll
    for (int v = 0; v < 8; ++v)
      D[(v + 8 * hi) * TSTR + t * 16 + mr] = acc[t][v] + bias[t * 16 + mr];
  __builtin_amdgcn_wave_barrier();
}

// In-place rowwise layernorm + silu on the wave's [16][TSTR] LDS tile.
__device__ __forceinline__ void wave_ln_silu(float* T, const float* g, const float* be,
                                             unsigned lane) {
  const int r = lane & 15;
  const int half = lane >> 4;
  float s = 0.f, s2 = 0.f;
  const float* row = T + r * TSTR + half * 32;
#pragma unroll
  for (int c = 0; c < 32; ++c) { float x = row[c]; s += x; s2 += x * x; }
  s += __shfl_xor(s, 16, 32);
  s2 += __shfl_xor(s2, 16, 32);
  float mu = s * (1.f / 64.f);
  float var = s2 * (1.f / 64.f) - mu * mu;
  float rstd = rsqrtf(var + 1e-5f);
#pragma unroll
  for (int c = 0; c < 32; ++c) {
    int cc = half * 32 + c;
    float x = (T[r * TSTR + cc] - mu) * rstd * g[cc] + be[cc];
    T[r * TSTR + cc] = x / (1.f + expf(-x));
  }
  __builtin_amdgcn_wave_barrier();
}

// OUT[16][dout] = A @ Wout + bias  (dout <= 16, Wto transposed [dout][WSTR])
__device__ __forceinline__ void wave_gemm_out_lds(const float* A, const float* Wto,
                                                  const float* bias, int dout, float* OUT,
                                                  int row0, int nrows,
                                                  const int* __restrict__ species,
                                                  unsigned lane) {
  const int mr = lane & 15;
  const int hi = lane >> 4;
  v8f acc = (v8f){};
#pragma unroll
  for (int k0 = 0; k0 < 64; k0 += 4) {
    const int kb = k0 + 2 * hi;
    v2f a = *(const v2f*)(A + mr * TSTR + kb);
    v2f b = (v2f){0.f, 0.f};
    if (mr < dout) b = *(const v2f*)(Wto + mr * WSTR + kb);
    acc = wmma_f32(a, b, acc);
  }
  if (mr < dout) {
#pragma unroll
    for (int v = 0; v < 8; ++v) {
      int m = v + 8 * hi;
      if (m < nrows) {
        int arow = row0 + m;
        float val = acc[v] + bias[mr];
        if (species) val *= (((float)species[arow]) > -0.5f) ? 1.f : 0.f;
        OUT[(size_t)arow * dout + mr] = val;
      }
    }
  }
}

__global__ __launch_bounds__(128)
void mlp64_wmma_kernel(const float* __restrict__ X, int n, const float* __restrict__ P,
                       int dout, float* __restrict__ OUT, const int* __restrict__ species) {
  __shared__ float sW0[64 * WSTR];          // W0 transposed: sW0[n*WSTR + k]
  __shared__ float sW1[64 * WSTR];
  __shared__ float sWo[16 * WSTR];          // Wout transposed (dout rows used)
  __shared__ float sVec[6 * 64 + 16];       // b0,g0,be0,b1,g1,be1,bout
  __shared__ float sT[4][16 * TSTR];        // per-wave activation tiles

  const int tid = threadIdx.x;

  // ---- cooperative param staging (coalesced global reads, transposed LDS) ----
  for (int idx = tid; idx < 4096; idx += 128) {
    int k = idx >> 6, nn = idx & 63;
    sW0[nn * WSTR + k] = P[idx];            // W0[k][n]
    sW1[nn * WSTR + k] = P[4288 + idx];     // W1[k][n]
  }
  for (int idx = tid; idx < 64 * dout; idx += 128) {
    int k = idx / dout, nn = idx % dout;
    sWo[nn * WSTR + k] = P[8576 + idx];     // Wout[k][n]
  }
  if (tid < 64) {
    sVec[tid]       = P[4096 + tid];        // b0
    sVec[64 + tid]  = P[4160 + tid];        // g0
    sVec[128 + tid] = P[4224 + tid];        // be0
    sVec[192 + tid] = P[8384 + tid];        // b1
    sVec[256 + tid] = P[8448 + tid];        // g1
    sVec[320 + tid] = P[8512 + tid];        // be1
  }
  if (tid < dout) sVec[384 + tid] = P[8576 + 64 * dout + tid];  // bout
  __syncthreads();

  const unsigned lane = tid & 31;
  const unsigned w = tid >> 5;
  const int row0 = (blockIdx.x * 4 + (int)w) * 16;
  if (row0 >= n) return;
  const int nrows = imin(16, n - row0);
  float* T = sT[w];

  // ---- stage this wave's X tile into LDS with coalesced b128 loads ----
#pragma unroll
  for (int i = 0; i < 8; ++i) {
    int idx = i * 32 + (int)lane;                 // 0..255 over [16 rows][16 float4]
    int r = idx >> 4, c4 = (idx & 15) * 4;
    int rr = (r < nrows) ? r : (nrows - 1);
    const float4 v = *(const float4*)(X + (size_t)(row0 + rr) * 64 + c4);
    *(float4*)(T + r * TSTR + c4) = v;
  }
  __builtin_amdgcn_wave_barrier();

  wave_gemm_lds(T, sW0, sVec + 0, T, lane);       // layer 0 (in-place)
  wave_ln_silu(T, sVec + 64, sVec + 128, lane);
  wave_gemm_lds(T, sW1, sVec + 192, T, lane);     // layer 1 (in-place)
  wave_ln_silu(T, sVec + 256, sVec + 320, lane);
  wave_gemm_out_lds(T, sWo, sVec + 384, dout, OUT, row0, nrows, species, lane);
}

// ---------------------------------------------------------------------------
extern "C" void kernel_launch(void* const* d_in, const int* in_sizes, int n_in,
                              void* d_out, int out_size, void* d_ws, size_t ws_size,
                              hipStream_t stream) {
  const float* cart    = (const float*)d_in[0];
  const int*   nl      = (const int*)d_in[1];
  const float* shifts  = (const float*)d_in[2];
  const int*   species = (const int*)d_in[3];
  const float* ccoef   = (const float*)d_in[4];
  const float* embN    = (const float*)d_in[5];
  const float* embC    = (const float*)d_in[6];
  const float* ocP     = (const float*)d_in[7];
  const float* outP    = (const float*)d_in[8];

  const int n_atom = in_sizes[0] / 3;
  const int n_edge = in_sizes[2] / 3;

  float* Wks    = (float*)d_ws;
  float* consts = Wks;                                   // [0..23] neigh_emb, [32..95] cc64
  float* coA    = Wks + 128;                             // n_atom*32
  float* coB    = coA + (size_t)n_atom * 32;             // n_atom*32
  float* dens   = coB + (size_t)n_atom * 32;             // n_atom*64
  float* iterc  = dens + (size_t)n_atom * 64;            // n_atom*8

  float* dvec = (float*)d_out;                           // [n_edge*3]
  float* outv = dvec + (size_t)n_edge * 3;               // [n_atom]

  nn_consts_kernel<<<1, 1, 0, stream>>>(embN, embC, consts);

  hipMemsetAsync(coA, 0, (size_t)n_atom * 32 * sizeof(float), stream);
  const int eb = (n_edge + 255) / 256;
  edge_setup_kernel<<<eb, 256, 0, stream>>>(cart, nl, shifts, consts, dvec, coA, n_edge);

  const int db = (n_atom * 64 + 255) / 256;
  density_kernel<<<db, 256, 0, stream>>>(coA, ccoef, consts + 32, dens, n_atom);

  const int tiles = (n_atom + 15) / 16;
  const int mb = (tiles + 3) / 4;

  float* cur = coA;
  float* nxt = coB;
  for (int i = 0; i < 3; ++i) {
    mlp64_wmma_kernel<<<mb, 128, 0, stream>>>(dens, n_atom, ocP + (size_t)i * 9096, 8, iterc, nullptr);
    hipMemsetAsync(nxt, 0, (size_t)n_atom * 32 * sizeof(float), stream);
    oc_edge_kernel<<<eb, 256, 0, stream>>>(dvec, nl, consts, iterc, cur, nxt, n_edge);
    density_kernel<<<db, 256, 0, stream>>>(nxt, ccoef + (size_t)(i + 1) * 1024, consts + 32, dens, n_atom);
    float* t = cur; cur = nxt; nxt = t;
  }

  mlp64_wmma_kernel<<<mb, 128, 0, stream>>>(dens, n_atom, outP, 1, outv, species);
}